// FrustumSampler_50354196578763
// MI455X (gfx1250) — compile-verified
//
#include <hip/hip_runtime.h>

// Inverse trilinear splat: scatter-add ray-sample features into a 256^3 volume.
// B=2, R=16384, P=256, C=1. Output (B,1,256,256,256) float32 = 134 MB, which
// fits in MI455X's 192 MB L2 -> global_atomic_add_f32 resolves in L2.
// One block = one ray (256 samples, 8 wave32s). Per-ray lengths/features are
// staged into LDS via the gfx1250 async global->LDS path (ASYNCcnt).

constexpr int kVol = 256;
constexpr int kDHW = kVol * kVol * kVol;
constexpr int kP   = 256;   // samples per ray == blockDim.x

__global__ void vol_zero_kernel(float4* __restrict__ p, int n4) {
    int i = blockIdx.x * blockDim.x + threadIdx.x;
    if (i < n4) p[i] = make_float4(0.f, 0.f, 0.f, 0.f);
}

#if defined(__HIP_DEVICE_COMPILE__) && __has_builtin(__builtin_amdgcn_global_load_async_to_lds_b128)
#define SPLAT_ASYNC_LDS 1
typedef int v4i __attribute__((ext_vector_type(4)));
typedef __attribute__((address_space(1))) v4i* global_v4i_ptr;
typedef __attribute__((address_space(3))) v4i* lds_v4i_ptr;
#else
#define SPLAT_ASYNC_LDS 0
#endif

__launch_bounds__(kP)
__global__ void splat_kernel(const float* __restrict__ origins,
                             const float* __restrict__ directions,
                             const float* __restrict__ lengths,
                             const float* __restrict__ feats,
                             float* __restrict__ out,
                             int raysPerBatch) {
    const int rid = blockIdx.x;       // global ray id in [0, B*R)
    const int tid = threadIdx.x;      // sample id in [0, P)
    const int b   = rid / raysPerBatch;

    // Block-uniform ray parameters (compiler emits scalar loads).
    // x = (ox + dx*t + 1)*128 - 0.5  ==  (dx*128)*t + (ox*128 + 127.5)
    const float ax = directions[3 * rid + 0] * 128.f;
    const float ay = directions[3 * rid + 1] * 128.f;
    const float az = directions[3 * rid + 2] * 128.f;
    const float bx = fmaf(origins[3 * rid + 0], 128.f, 127.5f);
    const float by = fmaf(origins[3 * rid + 1], 128.f, 127.5f);
    const float bz = fmaf(origins[3 * rid + 2], 128.f, 127.5f);

    const float* __restrict__ glen  = lengths + (size_t)rid * kP;
    const float* __restrict__ gfeat = feats   + (size_t)rid * kP;

    float t, f;
#if SPLAT_ASYNC_LDS
    __shared__ __align__(16) float s_len[kP];
    __shared__ __align__(16) float s_feat[kP];
    // 64 lanes x b128 covers each 256-float array (1 KB each).
    if (tid < kP / 4) {
        __builtin_amdgcn_global_load_async_to_lds_b128(
            (global_v4i_ptr)(glen + 4 * tid),
            (lds_v4i_ptr)(s_len + 4 * tid),
            0, 0);
    } else if (tid < kP / 2) {
        const int k = tid - kP / 4;
        __builtin_amdgcn_global_load_async_to_lds_b128(
            (global_v4i_ptr)(gfeat + 4 * k),
            (lds_v4i_ptr)(s_feat + 4 * k),
            0, 0);
    }
#if __has_builtin(__builtin_amdgcn_s_wait_asynccnt)
    __builtin_amdgcn_s_wait_asynccnt(0);
#else
    asm volatile("s_wait_asynccnt 0" ::: "memory");
#endif
    __syncthreads();
    t = s_len[tid];
    f = s_feat[tid];
#else
    t = glen[tid];
    f = gfeat[tid];
#endif

    const float x = fmaf(ax, t, bx);
    const float y = fmaf(ay, t, by);
    const float z = fmaf(az, t, bz);
    const float xf = floorf(x), yf = floorf(y), zf = floorf(z);
    const float fx = x - xf, fy = y - yf, fz = z - zf;
    const int x0 = (int)xf, y0 = (int)yf, z0 = (int)zf;

    const float wx[2] = {1.f - fx, fx};
    const float wy[2] = {1.f - fy, fy};
    const float wz[2] = {1.f - fz, fz};

    float* __restrict__ outb = out + (size_t)b * kDHW;

#pragma unroll
    for (int c = 0; c < 8; ++c) {
        const int dx = c & 1;
        const int dy = (c >> 1) & 1;
        const int dz = c >> 2;
        const unsigned xi = (unsigned)(x0 + dx);
        const unsigned yi = (unsigned)(y0 + dy);
        const unsigned zi = (unsigned)(z0 + dz);
        // Out-of-bounds corners contribute exactly 0 in the reference -> skip.
        if ((xi < 256u) & (yi < 256u) & (zi < 256u)) {
            const float w = wx[dx] * wy[dy] * wz[dz];
            atomicAdd(outb + ((((size_t)zi << 8) | yi) << 8 | xi), w * f);
        }
    }
}

extern "C" void kernel_launch(void* const* d_in, const int* in_sizes, int n_in,
                              void* d_out, int out_size, void* d_ws, size_t ws_size,
                              hipStream_t stream) {
    const float* origins    = (const float*)d_in[0];
    const float* directions = (const float*)d_in[1];
    const float* lengths    = (const float*)d_in[2];
    const float* feats      = (const float*)d_in[3];
    float* out = (float*)d_out;

    const int numRays = in_sizes[0] / 3;                 // B*R = 32768
    int B = out_size / kDHW;                             // 2
    if (B < 1) B = 1;
    const int raysPerBatch = numRays / B;                // R = 16384

    // Atomic accumulation target must be zeroed on every launch.
    const int n4 = out_size / 4;                         // out_size divisible by 4
    vol_zero_kernel<<<(n4 + 255) / 256, 256, 0, stream>>>((float4*)out, n4);

    splat_kernel<<<numRays, kP, 0, stream>>>(origins, directions, lengths,
                                             feats, out, raysPerBatch);
}